// HeterogeneousNetwork_27573690040479
// MI455X (gfx1250) — compile-verified
//
#include <hip/hip_runtime.h>

// ---------------------------------------------------------------------------
// Heterogeneous GraphSAGE (3 layers) for MI455X / gfx1250.
//   - edge scatter: GLOBAL_ATOMIC_ADD_F32 into L2-resident msg buffers
//   - GEMMs: V_WMMA_F32_16X16X4_F32, one 16x16 C tile per wave32,
//     branchless inner loop (W pre-transposed & zero-padded; row scale,
//     bias, ReLU and C-accumulation fused into the epilogue)
// ---------------------------------------------------------------------------

#define N_ROOMS   20000
#define N_OBJECTS 100000
#define IN_DIM    300
#define HID       128
#define OUT_DIM   25

typedef __attribute__((ext_vector_type(2))) float v2f;
typedef __attribute__((ext_vector_type(8))) float v8f;

__device__ __forceinline__ void atomic_add_f32(float* p, float v) {
  // lowers to GLOBAL_ATOMIC_ADD_F32 (no return)
  __hip_atomic_fetch_add(p, v, __ATOMIC_RELAXED, __HIP_MEMORY_SCOPE_AGENT);
}

// ---- degree counting -------------------------------------------------------
__global__ void k_count(const int* __restrict__ dst, int ne, float* __restrict__ deg) {
  int t = blockIdx.x * blockDim.x + threadIdx.x;
  if (t < ne) atomic_add_f32(&deg[dst[t]], 1.0f);
}

__global__ void k_inv(float* __restrict__ deg, int n) {
  int t = blockIdx.x * blockDim.x + threadIdx.x;
  if (t < n) deg[t] = 1.0f / fmaxf(deg[t], 1.0f);
}

// ---- weight prep: Wt[j][k] = (j<N ? Wa[k][j] (+ Wb[k][j]) : 0), j<N16 ------
// Transpose to column-major + pad columns to a multiple of 16 so the GEMM
// inner loop is branchless and B loads are contiguous b64 per lane.
__global__ void k_prepw(const float* __restrict__ Wa, const float* __restrict__ Wb,
                        float* __restrict__ Wt, int K, int N, int N16) {
  int t = blockIdx.x * blockDim.x + threadIdx.x;
  if (t >= N16 * K) return;
  int j = t / K;
  int k = t - j * K;
  float v = 0.0f;
  if (j < N) {
    v = Wa[k * N + j];
    if (Wb) v += Wb[k * N + j];
  }
  Wt[j * K + k] = v;
}

// ---- edge scatter: msg[dst] += x[src] (float4 granularity) -----------------
__global__ void k_scatter(const float* __restrict__ x, const int* __restrict__ src,
                          const int* __restrict__ dst, float* __restrict__ msg,
                          int ne, int d) {
  int nchunk = d >> 2;                         // d is a multiple of 4 (300, 128)
  int t = blockIdx.x * blockDim.x + threadIdx.x;
  if (t >= ne * nchunk) return;
  int e = t / nchunk;
  int c = t - e * nchunk;                      // consecutive threads -> same row, coalesced
  int s = src[e];
  int dd = dst[e];
  const float4 v = *(const float4*)(x + s * d + (c << 2));
  float* m = msg + dd * d + (c << 2);
  atomic_add_f32(m + 0, v.x);
  atomic_add_f32(m + 1, v.y);
  atomic_add_f32(m + 2, v.z);
  atomic_add_f32(m + 3, v.w);
}

// ---- WMMA GEMM: C (+)= diag(scale) * A[MxK] * Wt^T (+ bias) (+ ReLU) -------
// Wt is [N16][K] (transposed, zero-padded). One wave32 per 16x16 C tile.
// M multiple of 16, K multiple of 4. ldc may differ from N16 (e.g. 25).
__global__ void k_gemm(const float* __restrict__ A, const float* __restrict__ Wt,
                       const float* __restrict__ scale, const float* __restrict__ bias,
                       float* __restrict__ C, int M, int K, int N, int ldc,
                       int init, int relu) {
  int tiles_n = (N + 15) >> 4;
  int tiles_m = M >> 4;
  int tid = blockIdx.x * 4 + (threadIdx.x >> 5);
  if (tid >= tiles_m * tiles_n) return;
  int tn = tid % tiles_n;
  int tm = tid / tiles_n;

  int lane = threadIdx.x & 31;
  int l16  = lane & 15;
  int hi   = lane >> 4;                        // 0: K sub {0,1}, 1: K sub {2,3}

  int row = tm * 16 + l16;                     // A row held by this lane
  int col = tn * 16 + l16;                     // B/C column held by this lane

  const float* Arow = A + row * K + 2 * hi;    // per-lane A stream
  const float* Bcol = Wt + col * K + 2 * hi;   // per-lane B stream (contiguous)

  v8f c = (v8f){0.f, 0.f, 0.f, 0.f, 0.f, 0.f, 0.f, 0.f};

  // Branchless, full-EXEC inner loop: 2x b64 load + 1x wmma.
  for (int k = 0; k < K; k += 4) {
    v2f a = *(const v2f*)(Arow + k);           // A[row][k+2*hi .. +1]
    v2f b = *(const v2f*)(Bcol + k);           // W [k+2*hi .. +1][col]
    c = __builtin_amdgcn_wmma_f32_16x16x4_f32(false, a, false, b,
                                              (short)0, c, false, false);
  }

  // Epilogue: out = C_prev + scale[row]*acc + bias, optional ReLU.
  if (col < N) {
    float bv = bias ? bias[col] : 0.0f;
    int r0 = tm * 16 + 8 * hi;                 // first C row held by this lane
#pragma unroll
    for (int v = 0; v < 8; ++v) {
      float o = c[v];
      if (scale) o *= scale[r0 + v];
      o += bv;
      if (!init) o += C[(r0 + v) * ldc + col];
      if (relu) o = fmaxf(o, 0.0f);
      C[(r0 + v) * ldc + col] = o;
    }
  }
}

// ---------------------------------------------------------------------------
static inline int cdiv(int a, int b) { return (a + b - 1) / b; }

extern "C" void kernel_launch(void* const* d_in, const int* in_sizes, int n_in,
                              void* d_out, int out_size, void* d_ws, size_t ws_size,
                              hipStream_t stream) {
  (void)n_in; (void)out_size; (void)ws_size;

  const float* x_rooms   = (const float*)d_in[0];
  const float* x_objects = (const float*)d_in[1];
  const int* src_rr = (const int*)d_in[2]; const int* dst_rr = (const int*)d_in[3];
  const int* src_oo = (const int*)d_in[4]; const int* dst_oo = (const int*)d_in[5];
  const int* src_ro = (const int*)d_in[6]; const int* dst_ro = (const int*)d_in[7];
  const int* src_or = (const int*)d_in[8]; const int* dst_or = (const int*)d_in[9];
  const int ne_rr = in_sizes[2], ne_oo = in_sizes[4], ne_ro = in_sizes[6], ne_or = in_sizes[8];

  // params flattened in insertion order: layer{0,1,2} x et{rr,oo,ro,or} x {Wl,Wr,b}
  enum { ET_RR = 0, ET_OO = 1, ET_RO = 2, ET_OR = 3 };
  auto P = [&](int layer, int et, int which) -> const float* {
    return (const float*)d_in[10 + layer * 12 + et * 3 + which];
  };

  // ---- workspace layout (floats) ----
  float* ws     = (float*)d_ws;
  float* msg_r  = ws;                                 // 20000*300
  float* msg_o  = msg_r + N_ROOMS * IN_DIM;           // 100000*300
  float* deg    = msg_o + N_OBJECTS * IN_DIM;         // 240000
  float* inv_rr = deg;
  float* inv_or = deg + N_ROOMS;
  float* inv_oo = deg + 2 * N_ROOMS;
  float* inv_ro = deg + 2 * N_ROOMS + N_OBJECTS;
  float* h_ra   = deg + 2 * N_ROOMS + 2 * N_OBJECTS;  // 20000*128
  float* h_oa   = h_ra + N_ROOMS * HID;               // 100000*128
  float* h_rb   = h_oa + N_OBJECTS * HID;             // 20000*128
  float* h_ob   = h_rb + N_ROOMS * HID;               // 100000*128
  float* wt     = h_ob + N_OBJECTS * HID;             // 300*128 (padded/transposed W)

  auto prepw = [&](const float* a, const float* b, int K, int N) {
    int n16 = (N + 15) & ~15;
    k_prepw<<<cdiv(n16 * K, 256), 256, 0, stream>>>(a, b, wt, K, N, n16);
  };
  auto gemm = [&](const float* A, const float* scale, const float* bias,
                  float* C, int M, int K, int N, int ldc, int init, int relu) {
    int tiles = (M >> 4) * ((N + 15) >> 4);
    k_gemm<<<cdiv(tiles, 4), 128, 0, stream>>>(A, wt, scale, bias, C,
                                               M, K, N, ldc, init, relu);
  };
  auto scatter = [&](const float* x, const int* src, const int* dst, float* msg,
                     int ne, int d) {
    int total = ne * (d >> 2);
    k_scatter<<<cdiv(total, 256), 256, 0, stream>>>(x, src, dst, msg, ne, d);
  };

  // ---- degrees -> 1/max(deg,1) (edge-only, computed once per call) ----
  hipMemsetAsync(deg, 0, (2 * N_ROOMS + 2 * N_OBJECTS) * sizeof(float), stream);
  k_count<<<cdiv(ne_rr, 256), 256, 0, stream>>>(dst_rr, ne_rr, inv_rr);
  k_count<<<cdiv(ne_or, 256), 256, 0, stream>>>(dst_or, ne_or, inv_or);
  k_count<<<cdiv(ne_oo, 256), 256, 0, stream>>>(dst_oo, ne_oo, inv_oo);
  k_count<<<cdiv(ne_ro, 256), 256, 0, stream>>>(dst_ro, ne_ro, inv_ro);
  k_inv<<<cdiv(2 * N_ROOMS + 2 * N_OBJECTS, 256), 256, 0, stream>>>(
      deg, 2 * N_ROOMS + 2 * N_OBJECTS);

  // one destination-type pass of one layer:
  //   out = x_dst @ (Wr_e1 + Wr_e2)
  //       + mean_e1 @ Wl_e1 + b_e1
  //       + mean_e2 @ Wl_e2 + b_e2          (+ ReLU)
  auto run_dst = [&](int layer, int K, int N, int ldc,
                     const float* x_dst, int n_dst, float* msg, float* out,
                     int et1, const float* x_s1, const int* s1, const int* d1, int ne1, const float* inv1,
                     int et2, const float* x_s2, const int* s2, const int* d2, int ne2, const float* inv2,
                     int relu) {
    prepw(P(layer, et1, 1), P(layer, et2, 1), K, N);           // Wr_e1 + Wr_e2
    gemm(x_dst, nullptr, nullptr, out, n_dst, K, N, ldc, /*init=*/1, 0);

    hipMemsetAsync(msg, 0, (size_t)n_dst * K * sizeof(float), stream);
    scatter(x_s1, s1, d1, msg, ne1, K);
    prepw(P(layer, et1, 0), nullptr, K, N);                    // Wl_e1
    gemm(msg, inv1, P(layer, et1, 2), out, n_dst, K, N, ldc, 0, 0);

    hipMemsetAsync(msg, 0, (size_t)n_dst * K * sizeof(float), stream);
    scatter(x_s2, s2, d2, msg, ne2, K);
    prepw(P(layer, et2, 0), nullptr, K, N);                    // Wl_e2
    gemm(msg, inv2, P(layer, et2, 2), out, n_dst, K, N, ldc, 0, relu);
  };

  // ---- layer 0: (300 -> 128), ReLU ----
  run_dst(0, IN_DIM, HID, HID, x_rooms, N_ROOMS, msg_r, h_ra,
          ET_RR, x_rooms,   src_rr, dst_rr, ne_rr, inv_rr,
          ET_OR, x_objects, src_or, dst_or, ne_or, inv_or, /*relu=*/1);
  run_dst(0, IN_DIM, HID, HID, x_objects, N_OBJECTS, msg_o, h_oa,
          ET_OO, x_objects, src_oo, dst_oo, ne_oo, inv_oo,
          ET_RO, x_rooms,   src_ro, dst_ro, ne_ro, inv_ro, /*relu=*/1);

  // ---- layer 1: (128 -> 128), ReLU ----
  run_dst(1, HID, HID, HID, h_ra, N_ROOMS, msg_r, h_rb,
          ET_RR, h_ra, src_rr, dst_rr, ne_rr, inv_rr,
          ET_OR, h_oa, src_or, dst_or, ne_or, inv_or, /*relu=*/1);
  run_dst(1, HID, HID, HID, h_oa, N_OBJECTS, msg_o, h_ob,
          ET_OO, h_oa, src_oo, dst_oo, ne_oo, inv_oo,
          ET_RO, h_ra, src_ro, dst_ro, ne_ro, inv_ro, /*relu=*/1);

  // ---- layer 2: (128 -> 25), rooms only (objects output unused), no ReLU ----
  run_dst(2, HID, OUT_DIM, OUT_DIM, h_rb, N_ROOMS, msg_r, (float*)d_out,
          ET_RR, h_rb, src_rr, dst_rr, ne_rr, inv_rr,
          ET_OR, h_ob, src_or, dst_or, ne_or, inv_or, /*relu=*/0);
}